// ContrastiveModel_54314156425467
// MI455X (gfx1250) — compile-verified
//
#include <hip/hip_runtime.h>
#include <stdint.h>

#define BB 8
#define DD 64
#define HH 56
#define WW 56
#define HWP (HH * WW)          // 3136
#define PP 1024
#define NNEG 10
#define KQ 4096
#define ROWS (BB * HWP)        // 25088
#define NCOLS (BB + KQ)        // 4104
#define INV_T (1.0f / 0.07f)

typedef __attribute__((ext_vector_type(2))) float v2f;
typedef __attribute__((ext_vector_type(8))) float v8f;

// ---------------------------------------------------------------------------
// Async global->LDS copy (CDNA5): per-lane 16B, tracked by ASYNCcnt.
// VDST = LDS byte address VGPR, VADDR = 64-bit global address VGPR pair.
// ---------------------------------------------------------------------------
__device__ __forceinline__ void async_g2l_b128(uint32_t lds_addr, const float* gaddr) {
    asm volatile("global_load_async_to_lds_b128 %0, %1, off"
                 :: "v"(lds_addr), "v"(gaddr)
                 : "memory");
}
__device__ __forceinline__ void wait_async0() {
    asm volatile("s_wait_asynccnt 0" ::: "memory");
}

// ---------------------------------------------------------------------------
// Kernel 1: per-pixel L2 normalize of q and k.
//   qn: [B*HW, 64]  (pixel-major: GEMM A operand + gather layout)
//   kn: [B, 64, HW] (channel-major: pooling / proto / neg-column layout)
// ---------------------------------------------------------------------------
__global__ void moco_normalize_kernel(const float* __restrict__ q,
                                      const float* __restrict__ k,
                                      float* __restrict__ qn,
                                      float* __restrict__ kn) {
    int i = blockIdx.x * blockDim.x + threadIdx.x;
    if (i >= ROWS) return;
    int b = i / HWP;
    int hw = i - b * HWP;
    const float* qp = q + (size_t)b * DD * HWP + hw;
    const float* kp = k + (size_t)b * DD * HWP + hw;
    float sq = 0.f, sk = 0.f;
    #pragma unroll 8
    for (int d = 0; d < DD; ++d) {
        float v = qp[(size_t)d * HWP]; sq += v * v;
        float w = kp[(size_t)d * HWP]; sk += w * w;
    }
    float rq = 1.f / fmaxf(sqrtf(sq), 1e-12f);
    float rk = 1.f / fmaxf(sqrtf(sk), 1e-12f);
    float* qo = qn + (size_t)i * DD;
    float* ko = kn + (size_t)b * DD * HWP + hw;
    #pragma unroll 8
    for (int d = 0; d < DD; ++d) {
        qo[d]                 = qp[(size_t)d * HWP] * rq;
        ko[(size_t)d * HWP]   = kp[(size_t)d * HWP] * rk;
    }
}

// ---------------------------------------------------------------------------
// Kernel 2: saliency-weighted prototype per image, L2-normalized over D.
// One block per image, 64 threads (one per channel).
// ---------------------------------------------------------------------------
__global__ void moco_protos_kernel(const float* __restrict__ kn,
                                   const int* __restrict__ sal,
                                   float* __restrict__ protos) {
    int b = blockIdx.x;
    int d = threadIdx.x;
    const float* kd = kn + ((size_t)b * DD + d) * HWP;
    const int* sp = sal + (size_t)b * HWP;
    float acc = 0.f;
    for (int hw = 0; hw < HWP; ++hw)
        acc += kd[hw] * (float)sp[hw];
    __shared__ float sh[DD];
    sh[d] = acc * acc;
    __syncthreads();
    if (d == 0) {
        float s = 0.f;
        #pragma unroll
        for (int t = 0; t < DD; ++t) s += sh[t];
        sh[0] = fmaxf(sqrtf(s), 1e-12f);
    }
    __syncthreads();
    protos[b * DD + d] = acc / sh[0];
}

// ---------------------------------------------------------------------------
// Kernel 3: first 8 columns of logits = q_flat @ protos^T / T.
// ---------------------------------------------------------------------------
__global__ void moco_proto_logits_kernel(const float* __restrict__ qn,
                                         const float* __restrict__ protos,
                                         float* __restrict__ out) {
    __shared__ float sp[BB * DD];
    for (int t = threadIdx.x; t < BB * DD; t += blockDim.x)
        sp[t] = protos[t];
    __syncthreads();
    int row = blockIdx.x * blockDim.x + threadIdx.x;
    if (row >= ROWS) return;
    const float* qr = qn + (size_t)row * DD;
    #pragma unroll
    for (int bb = 0; bb < BB; ++bb) {
        float acc = 0.f;
        #pragma unroll 8
        for (int d = 0; d < DD; ++d) acc += qr[d] * sp[bb * DD + d];
        out[(size_t)row * NCOLS + bb] = acc * INV_T;
    }
}

// ---------------------------------------------------------------------------
// Kernel 4: main GEMM  q_flat[25088,64] @ queue[64,4096] / T  -> logits[:,8:]
// Block: 256 threads (8 waves), tile 32 rows x 128 cols.
// Wave: 16 rows x 32 cols = 2 x v_wmma_f32_16x16x4_f32 accumulators.
// A-tile + B-slab staged into LDS with global_load_async_to_lds_b128.
// LDS strides padded (A:68, B:136 floats) for conflict-free fragment reads.
// ---------------------------------------------------------------------------
#define A_STRIDE 68
#define B_STRIDE 136
__global__ __launch_bounds__(256) void moco_gemm_wmma_kernel(
        const float* __restrict__ qn,
        const float* __restrict__ queue,
        float* __restrict__ out) {
    __shared__ float ldsA[32 * A_STRIDE];   //  8.7 KB
    __shared__ float ldsB[64 * B_STRIDE];   // 34.8 KB
    const int tid = threadIdx.x;
    const int rb = blockIdx.x;              // 0..783  (row tile of 32)
    const int cb = blockIdx.y;              // 0..31   (col slab of 128)

    const uint32_t baseA = (uint32_t)(uintptr_t)ldsA;
    const uint32_t baseB = (uint32_t)(uintptr_t)ldsB;

    // Stage A: 32 rows x 64 floats = 512 float4, 2 per thread.
    #pragma unroll
    for (int j = 0; j < 2; ++j) {
        int idx = j * 256 + tid;
        int row = idx >> 4, seg = idx & 15;
        const float* g = qn + ((size_t)(rb * 32 + row)) * DD + seg * 4;
        async_g2l_b128(baseA + (uint32_t)(row * A_STRIDE + seg * 4) * 4u, g);
    }
    // Stage B: 64 k-rows x 128 cols = 2048 float4, 8 per thread.
    #pragma unroll
    for (int j = 0; j < 8; ++j) {
        int idx = j * 256 + tid;
        int row = idx >> 5, seg = idx & 31;
        const float* g = queue + (size_t)row * KQ + cb * 128 + seg * 4;
        async_g2l_b128(baseB + (uint32_t)(row * B_STRIDE + seg * 4) * 4u, g);
    }
    wait_async0();
    __syncthreads();

    const int lane = tid & 31;
    const int wid = tid >> 5;
    const int rowTile = wid & 1;        // which 16-row half
    const int colGroup = wid >> 1;      // 0..3 -> 32-col group
    const int rsel = lane >> 4;         // half-wave select
    const int l15 = lane & 15;

    v8f c0 = {};
    v8f c1 = {};
    const int arow = rowTile * 16 + l15;
    const int bcol = colGroup * 32 + l15;

    #pragma unroll
    for (int kk = 0; kk < 16; ++kk) {
        const int kb = kk * 4 + rsel * 2;     // this half-wave's K pair
        // A fragment: lane holds A[arow, kb], A[arow, kb+1]
        const float2 av = *(const float2*)&ldsA[arow * A_STRIDE + kb];
        v2f a; a.x = av.x; a.y = av.y;
        // B fragments: lane holds B[kb, col], B[kb+1, col]
        v2f b0, b1;
        b0.x = ldsB[kb * B_STRIDE + bcol];
        b0.y = ldsB[(kb + 1) * B_STRIDE + bcol];
        b1.x = ldsB[kb * B_STRIDE + bcol + 16];
        b1.y = ldsB[(kb + 1) * B_STRIDE + bcol + 16];
        c0 = __builtin_amdgcn_wmma_f32_16x16x4_f32(false, a, false, b0,
                                                   (short)0, c0, false, false);
        c1 = __builtin_amdgcn_wmma_f32_16x16x4_f32(false, a, false, b1,
                                                   (short)0, c1, false, false);
    }

    // C/D layout: VGPR r -> M = r + 8*rsel, N = l15.
    const int grow = rb * 32 + rowTile * 16 + rsel * 8;
    const int gcol = BB + cb * 128 + colGroup * 32 + l15;
    #pragma unroll
    for (int r = 0; r < 8; ++r) {
        size_t o = (size_t)(grow + r) * NCOLS + gcol;
        out[o]      = c0[r] * INV_T;
        out[o + 16] = c1[r] * INV_T;
    }
}

// ---------------------------------------------------------------------------
// Kernel 5: tmp + l_logits. One thread per foreground sample (b,p).
//   l_pos = <obj, avg3x3(kn)@idx>/T ; l_neg[j] = <obj, kn[:,neg_idx_j]>/T
// ---------------------------------------------------------------------------
__global__ void moco_llogits_kernel(const float* __restrict__ qn,
                                    const float* __restrict__ kn,
                                    const int* __restrict__ indexes,
                                    const int* __restrict__ neg_idx,
                                    float* __restrict__ tmp_out,
                                    float* __restrict__ ll_out) {
    int i = blockIdx.x * blockDim.x + threadIdx.x;
    if (i >= BB * PP) return;
    int b = i >> 10;
    tmp_out[i] = (float)b;

    int idx = indexes[i];
    const float* objp = qn + ((size_t)b * HWP + idx) * DD;
    float obj[DD];
    #pragma unroll 8
    for (int d = 0; d < DD; ++d) obj[d] = objp[d];

    const float* knb = kn + (size_t)b * DD * HWP;
    int r = idx / WW, c = idx - r * WW;

    // l_pos: 3x3 avg (count_include_pad -> /9) dotted with obj
    float lpos = 0.f;
    for (int d = 0; d < DD; ++d) {
        const float* kd = knb + (size_t)d * HWP;
        float s = 0.f;
        #pragma unroll
        for (int dr = -1; dr <= 1; ++dr) {
            int rr = r + dr;
            if (rr < 0 || rr >= HH) continue;
            #pragma unroll
            for (int dc = -1; dc <= 1; ++dc) {
                int cc = c + dc;
                if (cc < 0 || cc >= WW) continue;
                s += kd[rr * WW + cc];
            }
        }
        lpos += obj[d] * (s * (1.f / 9.f));
    }
    float* llr = ll_out + (size_t)i * (1 + NNEG);
    llr[0] = lpos * INV_T;

    #pragma unroll
    for (int j = 0; j < NNEG; ++j) {
        int col = neg_idx[(size_t)i * NNEG + j];
        const float* kc = knb + col;
        float acc = 0.f;
        #pragma unroll 8
        for (int d = 0; d < DD; ++d) acc += obj[d] * kc[(size_t)d * HWP];
        llr[1 + j] = acc * INV_T;
    }
}

// ---------------------------------------------------------------------------
extern "C" void kernel_launch(void* const* d_in, const int* in_sizes, int n_in,
                              void* d_out, int out_size, void* d_ws, size_t ws_size,
                              hipStream_t stream) {
    const float* q       = (const float*)d_in[0];
    const float* k       = (const float*)d_in[1];
    const float* queue   = (const float*)d_in[2];
    const int*   sal     = (const int*)d_in[3];
    const int*   indexes = (const int*)d_in[4];
    const int*   neg_idx = (const int*)d_in[5];

    float* out    = (float*)d_out;
    float* logits = out;                                    // [25088, 4104]
    float* tmp    = out + (size_t)ROWS * NCOLS;             // [8192]
    float* ll     = tmp + (size_t)BB * PP;                  // [8192, 11]

    float* qn     = (float*)d_ws;                           // [25088, 64]
    float* kn     = qn + (size_t)ROWS * DD;                 // [8, 64, 3136]
    float* protos = kn + (size_t)ROWS * DD;                 // [8, 64]

    moco_normalize_kernel<<<(ROWS + 255) / 256, 256, 0, stream>>>(q, k, qn, kn);
    moco_protos_kernel<<<BB, DD, 0, stream>>>(kn, sal, protos);
    moco_proto_logits_kernel<<<(ROWS + 255) / 256, 256, 0, stream>>>(qn, protos, logits);
    moco_gemm_wmma_kernel<<<dim3(ROWS / 32, KQ / 128), 256, 0, stream>>>(qn, queue, logits);
    moco_llogits_kernel<<<(BB * PP + 255) / 256, 256, 0, stream>>>(qn, kn, indexes,
                                                                   neg_idx, tmp, ll);
}